// RGCN_64536178589819
// MI455X (gfx1250) — compile-verified
//
#include <hip/hip_runtime.h>

#define N_NODES 50000
#define R_REL   4
#define E_EDGES 150000
#define F_FEAT  128   // IN == HID == 128
#define F_OUT   64
#define M_TILES (N_NODES / 16)   // 3125

typedef float v2f __attribute__((ext_vector_type(2)));
typedef float v8f __attribute__((ext_vector_type(8)));
typedef unsigned int u32x4 __attribute__((ext_vector_type(4)));
typedef int i32x4 __attribute__((ext_vector_type(4)));
typedef int i32x8 __attribute__((ext_vector_type(8)));

#if defined(__has_builtin)
#if __has_builtin(__builtin_amdgcn_tensor_load_to_lds) && __has_builtin(__builtin_amdgcn_s_wait_tensorcnt)
#define USE_TDM 1
#endif
#endif

// ---------------------------------------------------------------- utilities
__global__ void zero_kernel(float* __restrict__ p, int n) {
    int i = blockIdx.x * blockDim.x + threadIdx.x;
    if (i < n) p[i] = 0.0f;
}

__global__ void degree_kernel(const int* __restrict__ src, const int* __restrict__ dst,
                              float* __restrict__ deg_out, float* __restrict__ deg_in) {
    int i = blockIdx.x * blockDim.x + threadIdx.x;
    if (i >= R_REL * E_EDGES) return;
    int r = i / E_EDGES;
    atomicAdd(deg_out + r * N_NODES + src[i], 1.0f);
    atomicAdd(deg_in  + r * N_NODES + dst[i], 1.0f);
}

__global__ void norm_kernel(const float* __restrict__ deg_out, const float* __restrict__ deg_in,
                            float* __restrict__ norm_s, float* __restrict__ norm_d) {
    int i = blockIdx.x * blockDim.x + threadIdx.x;
    if (i >= R_REL * N_NODES) return;
    float o = deg_out[i];
    float d = deg_in[i];
    norm_s[i] = (o > 0.0f) ? rsqrtf(o) : 0.0f;   // degrees are integers
    norm_d[i] = (d > 0.0f) ? rsqrtf(d) : 0.0f;
}

// ------------------------------------------------------- scatter (message sum)
__global__ void scatter_kernel(const float* __restrict__ X, const int* __restrict__ src,
                               const int* __restrict__ dst, const float* __restrict__ norm_s,
                               float* __restrict__ agg) {
    int i = blockIdx.x * blockDim.x + threadIdx.x;
    if (i >= E_EDGES * (F_FEAT / 4)) return;
    int e = i >> 5;
    int c = (i & 31) << 2;
    int s = src[e];
    int d = dst[e];
    float ns = norm_s[s];
    float4 v = *reinterpret_cast<const float4*>(X + (size_t)s * F_FEAT + c);
    float* o = agg + (size_t)d * F_FEAT + c;
    atomicAdd(o + 0, v.x * ns);
    atomicAdd(o + 1, v.y * ns);
    atomicAdd(o + 2, v.z * ns);
    atomicAdd(o + 3, v.w * ns);
}

// ------------------------------------------------------------- WMMA GEMM
// acc[n, :COLS] += (agg[n,:] * norm_d[n]) @ W[128, COLS] + bias[COLS]
// Block: 4 waves, each wave one 16x64 tile (4 WMMA fragments, shared A fragment).
// W column-group [g*64, g*64+64) staged in 32KB LDS once per block (TDM if available).
template <int COLS>
__global__ void __launch_bounds__(128)
gemm_wmma_kernel(const float* __restrict__ A, const float* __restrict__ norm_d,
                 const float* __restrict__ W, const float* __restrict__ bias,
                 float* __restrict__ acc) {
    extern __shared__ float sW[];            // [128][64]
    const int lane = threadIdx.x & 31;
    const int wave = threadIdx.x >> 5;
    const int g    = blockIdx.y;             // 64-column group

#if USE_TDM
    if (wave == 0) {
        // Tensor DMA descriptor (ISA 8.3-8.6): 2D tile, 64 x 128 rows, f32.
        unsigned long long ga = (unsigned long long)(const void*)(W + (size_t)g * 64);
        unsigned lds_base = (unsigned)(unsigned long long)(void*)sW;  // flat low 32b = LDS offset
        u32x4 g0;
        g0[0] = 1u;                                        // count=1, user desc, no gather
        g0[1] = lds_base;                                  // lds_addr
        g0[2] = (unsigned)ga;                              // global_addr[31:0]
        g0[3] = (unsigned)((ga >> 32) & 0x1FFFFFFull) | (2u << 30);  // addr[56:32] | type=2
        i32x8 g1;
        g1[0] = (int)(2u << 16);                           // data_size = 4B
        g1[1] = (int)(64u << 16);                          // tensor_dim0 = 64  (bits 79:48)
        g1[2] = (int)(128u << 16);                         // tensor_dim1 = 128 (bits 111:80)
        g1[3] = (int)(64u << 16);                          // tile_dim0 = 64    (bits 127:112)
        g1[4] = (int)128u;                                 // tile_dim1 = 128, tile_dim2 = 0
        g1[5] = (int)(unsigned)COLS;                       // tensor_dim0_stride = COLS
        g1[6] = 0;
        g1[7] = 0;
        i32x4 g2 = {0, 0, 0, 0};                           // 2D: dims 2/3 unused
        i32x4 g3 = {0, 0, 0, 0};
        i32x8 g4 = {0, 0, 0, 0, 0, 0, 0, 0};               // unused trailing group (6-arg form)
        __builtin_amdgcn_tensor_load_to_lds(g0, g1, g2, g3, g4, 0);
        __builtin_amdgcn_s_wait_tensorcnt(0);
    }
#else
    // Cooperative staging: 128 threads x 16 float4 = 128x64 tile.
    for (int i = threadIdx.x; i < 128 * 16; i += 128) {
        int row = i >> 4;
        int c4  = (i & 15) << 2;
        *reinterpret_cast<float4*>(sW + row * 64 + c4) =
            *reinterpret_cast<const float4*>(W + (size_t)row * COLS + g * 64 + c4);
    }
#endif
    __syncthreads();

    const int tm = blockIdx.x * 4 + wave;
    if (tm >= M_TILES) return;               // wave-uniform; after the only barrier

    const int half = lane >> 4;              // 0: K+{0,1}, 1: K+{2,3}
    const int l16  = lane & 15;
    const int arow = tm * 16 + l16;
    const float nd = norm_d[arow];
    const float* __restrict__ Ap = A + (size_t)arow * F_FEAT + half * 2;
    const float* __restrict__ Bp = sW + half * 2 * 64 + l16;   // lane base into LDS tile

    v8f c0 = {}, c1 = {}, c2 = {}, c3 = {};
#pragma unroll
    for (int k = 0; k < F_FEAT; k += 4) {
        v2f a = *reinterpret_cast<const v2f*>(Ap + k);
        a[0] *= nd;
        a[1] *= nd;
        v2f b;
        b[0] = Bp[k * 64 +  0]; b[1] = Bp[k * 64 + 64 +  0];
        c0 = __builtin_amdgcn_wmma_f32_16x16x4_f32(false, a, false, b, (short)0, c0, false, false);
        b[0] = Bp[k * 64 + 16]; b[1] = Bp[k * 64 + 64 + 16];
        c1 = __builtin_amdgcn_wmma_f32_16x16x4_f32(false, a, false, b, (short)0, c1, false, false);
        b[0] = Bp[k * 64 + 32]; b[1] = Bp[k * 64 + 64 + 32];
        c2 = __builtin_amdgcn_wmma_f32_16x16x4_f32(false, a, false, b, (short)0, c2, false, false);
        b[0] = Bp[k * 64 + 48]; b[1] = Bp[k * 64 + 64 + 48];
        c3 = __builtin_amdgcn_wmma_f32_16x16x4_f32(false, a, false, b, (short)0, c3, false, false);
    }

    const int col0 = g * 64 + l16;
    const float bv0 = bias[col0 +  0];
    const float bv1 = bias[col0 + 16];
    const float bv2 = bias[col0 + 32];
    const float bv3 = bias[col0 + 48];
    float* __restrict__ accp = acc + (size_t)(tm * 16 + half * 8) * COLS + col0;
#pragma unroll
    for (int v = 0; v < 8; ++v) {            // C/D: VGPR v -> row v (+8 for hi lanes)
        accp[v * COLS +  0] += c0[v] + bv0;
        accp[v * COLS + 16] += c1[v] + bv1;
        accp[v * COLS + 32] += c2[v] + bv2;
        accp[v * COLS + 48] += c3[v] + bv3;
    }
}

__global__ void finalize_kernel(const float* __restrict__ acc, float* __restrict__ out,
                                int n, int do_relu) {
    int i = blockIdx.x * blockDim.x + threadIdx.x;
    if (i >= n) return;
    float v = acc[i] * (1.0f / R_REL);
    out[i] = do_relu ? fmaxf(v, 0.0f) : v;
}

// --------------------------------------------------------------------- host
extern "C" void kernel_launch(void* const* d_in, const int* in_sizes, int n_in,
                              void* d_out, int out_size, void* d_ws, size_t ws_size,
                              hipStream_t stream) {
    const float* x   = (const float*)d_in[0];
    const int*   src = (const int*)  d_in[1];
    const int*   dst = (const int*)  d_in[2];
    const float* Ws[3] = { (const float*)d_in[3], (const float*)d_in[5], (const float*)d_in[7] };
    const float* Bs[3] = { (const float*)d_in[4], (const float*)d_in[6], (const float*)d_in[8] };
    float* out = (float*)d_out;

    float* ws      = (float*)d_ws;
    float* deg_out = ws;
    float* deg_in  = deg_out + (size_t)R_REL * N_NODES;
    float* norm_s  = deg_in  + (size_t)R_REL * N_NODES;
    float* norm_d  = norm_s  + (size_t)R_REL * N_NODES;
    float* hA      = norm_d  + (size_t)R_REL * N_NODES;
    float* hB      = hA      + (size_t)N_NODES * F_FEAT;
    float* agg     = hB      + (size_t)N_NODES * F_FEAT;
    float* acc     = agg     + (size_t)N_NODES * F_FEAT;

    auto zero = [&](float* p, size_t n) {
        zero_kernel<<<(unsigned)((n + 255) / 256), 256, 0, stream>>>(p, (int)n);
    };

    zero(deg_out, (size_t)2 * R_REL * N_NODES);
    degree_kernel<<<(R_REL * E_EDGES + 255) / 256, 256, 0, stream>>>(src, dst, deg_out, deg_in);
    norm_kernel<<<(R_REL * N_NODES + 255) / 256, 256, 0, stream>>>(deg_out, deg_in, norm_s, norm_d);

    const float* Xs[3]   = { x, hA, hB };
    float*       Os[3]   = { hA, hB, out };
    const int    colsA[3] = { F_FEAT, F_FEAT, F_OUT };
    const unsigned mblocks = (M_TILES + 3) / 4;   // 782

    for (int l = 0; l < 3; ++l) {
        const int cols = colsA[l];
        zero(acc, (size_t)N_NODES * cols);
        for (int r = 0; r < R_REL; ++r) {
            zero(agg, (size_t)N_NODES * F_FEAT);
            scatter_kernel<<<(E_EDGES * (F_FEAT / 4) + 255) / 256, 256, 0, stream>>>(
                Xs[l], src + (size_t)r * E_EDGES, dst + (size_t)r * E_EDGES,
                norm_s + (size_t)r * N_NODES, agg);
            if (cols == F_FEAT) {
                gemm_wmma_kernel<F_FEAT><<<dim3(mblocks, 2), 128, 32768, stream>>>(
                    agg, norm_d + (size_t)r * N_NODES,
                    Ws[l] + (size_t)r * F_FEAT * cols, Bs[l] + (size_t)r * cols, acc);
            } else {
                gemm_wmma_kernel<F_OUT><<<dim3(mblocks, 1), 128, 32768, stream>>>(
                    agg, norm_d + (size_t)r * N_NODES,
                    Ws[l] + (size_t)r * F_FEAT * cols, Bs[l] + (size_t)r * cols, acc);
            }
        }
        finalize_kernel<<<((size_t)N_NODES * cols + 255) / 256, 256, 0, stream>>>(
            acc, Os[l], N_NODES * cols, (l < 2) ? 1 : 0);
    }
}